// Encoder_17454747091291
// MI455X (gfx1250) — compile-verified
//
#include <hip/hip_runtime.h>

// ---- problem constants (match reference) ----
#define IN_F   128
#define HID_F  128   // 2*H
#define OUT_F  64    // H
#define RRELU_SLOPE ((1.0f/8.0f + 1.0f/3.0f) * 0.5f)

typedef float v2f __attribute__((ext_vector_type(2)));
typedef float v8f __attribute__((ext_vector_type(8)));

// ------------------------------------------------------------------
// degree / normalization
// ------------------------------------------------------------------
__global__ void k_init_deg(float* __restrict__ deg, int n) {
  int i = blockIdx.x * blockDim.x + threadIdx.x;
  if (i < n) deg[i] = 1.0f;            // self-loop contributes 1
}

__global__ void k_accum_deg(const int* __restrict__ col, float* __restrict__ deg, int e) {
  int i = blockIdx.x * blockDim.x + threadIdx.x;
  if (i < e) unsafeAtomicAdd(&deg[col[i]], 1.0f);
}

__global__ void k_dinv(float* __restrict__ deg, int n) {
  int i = blockIdx.x * blockDim.x + threadIdx.x;
  if (i < n) {
    float d = deg[i];
    deg[i] = (d > 0.0f) ? rsqrtf(d) : 0.0f;   // in place: deg -> dinv
  }
}

// ------------------------------------------------------------------
// GEMM: Y[n x KOUT] = X[n x KIN] @ W[KIN x KOUT]  (fp32 WMMA 16x16x4)
// one 16x16 output tile per wave; 4 waves / block
// ------------------------------------------------------------------
template<int KIN, int KOUT>
__global__ void k_gemm_wmma(const float* __restrict__ X,
                            const float* __restrict__ W,
                            float* __restrict__ Y,
                            int nRows) {
  const int lane  = threadIdx.x & 31;
  const int wave  = threadIdx.x >> 5;
  const int nTiles = KOUT / 16;
  const int mTiles = nRows / 16;           // nRows divisible by 16 (100000)
  int tile = blockIdx.x * 4 + wave;
  if (tile >= mTiles * nTiles) return;     // wave-uniform: EXEC stays all-ones
  const int mTile = tile / nTiles;
  const int nTile = tile % nTiles;

  const int half = lane >> 4;              // 0: K=0,1 | 1: K=2,3 slice
  const int l16  = lane & 15;
  const int m = mTile * 16 + l16;          // A row owned by this lane
  const int n = nTile * 16 + l16;          // B/D column owned by this lane

  const float* __restrict__ xrow = X + (size_t)m * KIN;

  v8f c = {};
#pragma unroll 4
  for (int k0 = 0; k0 < KIN; k0 += 4) {
    const int ka = k0 + half * 2;
    v2f a;
    a.x = xrow[ka];
    a.y = xrow[ka + 1];
    v2f b;
    b.x = W[(size_t)ka       * KOUT + n];
    b.y = W[(size_t)(ka + 1) * KOUT + n];
    c = __builtin_amdgcn_wmma_f32_16x16x4_f32(
            /*neg_a=*/false, a, /*neg_b=*/false, b,
            /*c_mod=*/(short)0, c, /*reuse_a=*/false, /*reuse_b=*/false);
  }

  // D layout: VGPR v -> row mTile*16 + v + 8*half, col n
  float* __restrict__ yb = Y + ((size_t)(mTile * 16 + half * 8)) * KOUT + nTile * 16 + l16;
#pragma unroll
  for (int v = 0; v < 8; ++v)
    yb[(size_t)v * KOUT] = c[v];
}

// ------------------------------------------------------------------
// out[i][f] = bias[f] + h[i][f] * dinv[i]^2     (self-loop + bias init)
// ------------------------------------------------------------------
template<int F>
__global__ void k_self_init(const float* __restrict__ h,
                            const float* __restrict__ dinv,
                            const float* __restrict__ bias,
                            float* __restrict__ out, int n) {
  int idx = blockIdx.x * blockDim.x + threadIdx.x;
  if (idx < n * F) {
    int i = idx / F;
    int f = idx & (F - 1);
    float di = dinv[i];
    out[idx] = bias[f] + h[idx] * di * di;
  }
}

// ------------------------------------------------------------------
// edge scatter: out[col[e]] += h[row[e]] * dinv[row]*dinv[col]
// one wave per edge; each lane owns F/32 contiguous floats
// ------------------------------------------------------------------
template<int F>
__global__ void k_edge_scatter(const int* __restrict__ rowIdx,
                               const int* __restrict__ colIdx,
                               const float* __restrict__ dinv,
                               const float* __restrict__ h,
                               float* __restrict__ out, int e) {
  const int lane = threadIdx.x & 31;
  const long ed  = (long)blockIdx.x * (blockDim.x >> 5) + (threadIdx.x >> 5);
  if (ed >= e) return;
  const int r = rowIdx[ed];
  const int c = colIdx[ed];
  const float nrm = dinv[r] * dinv[c];
  constexpr int V = F / 32;
  const float* __restrict__ hs = h   + (size_t)r * F + lane * V;
  float*       __restrict__ od = out + (size_t)c * F + lane * V;
#pragma unroll
  for (int i = 0; i < V; ++i)
    unsafeAtomicAdd(&od[i], hs[i] * nrm);
}

// ------------------------------------------------------------------
// rrelu (eval): x >= 0 ? x : slope * x   (in place)
// ------------------------------------------------------------------
__global__ void k_rrelu(float* __restrict__ x, long n) {
  long i = (long)blockIdx.x * blockDim.x + threadIdx.x;
  if (i < n) {
    float v = x[i];
    x[i] = (v >= 0.0f) ? v : (RRELU_SLOPE * v);
  }
}

// ------------------------------------------------------------------
// launch
// ------------------------------------------------------------------
extern "C" void kernel_launch(void* const* d_in, const int* in_sizes, int n_in,
                              void* d_out, int out_size, void* d_ws, size_t ws_size,
                              hipStream_t stream) {
  const float* x    = (const float*)d_in[0];
  const int*   ei   = (const int*)  d_in[1];
  const float* W1   = (const float*)d_in[2];
  const float* b1   = (const float*)d_in[3];
  const float* W2   = (const float*)d_in[4];
  const float* b2   = (const float*)d_in[5];
  float*       out  = (float*)d_out;

  const int N = in_sizes[0] / IN_F;
  const int E = in_sizes[1] / 2;
  const int* row = ei;       // edge_index[0]
  const int* col = ei + E;   // edge_index[1]

  // workspace layout: dinv[N] | bufA[N*128] | bufB[N*128]
  float* dinv = (float*)d_ws;
  float* bufA = dinv + N;                 // h1 (layer1 GEMM out), later h2
  float* bufB = bufA + (size_t)N * HID_F; // g1 (layer1 aggregated)

  const int T = 256;

  // degrees -> dinv
  k_init_deg <<<(N + T - 1) / T, T, 0, stream>>>(dinv, N);
  k_accum_deg<<<(E + T - 1) / T, T, 0, stream>>>(col, dinv, E);
  k_dinv     <<<(N + T - 1) / T, T, 0, stream>>>(dinv, N);

  // ---- layer 1: h1 = x @ W1 ; g1 = scatter(norm * h1[row]) + b1 ; rrelu
  {
    const int tiles  = (N / 16) * (HID_F / 16);
    k_gemm_wmma<IN_F, HID_F><<<(tiles + 3) / 4, 128, 0, stream>>>(x, W1, bufA, N);

    const long elems = (long)N * HID_F;
    k_self_init<HID_F><<<(int)((elems + T - 1) / T), T, 0, stream>>>(bufA, dinv, b1, bufB, N);

    const int wavesPerBlk = T / 32;
    k_edge_scatter<HID_F><<<(E + wavesPerBlk - 1) / wavesPerBlk, T, 0, stream>>>(
        row, col, dinv, bufA, bufB, E);

    k_rrelu<<<(int)((elems + T - 1) / T), T, 0, stream>>>(bufB, elems);
  }

  // ---- layer 2: h2 = g1 @ W2 ; out = scatter(norm * h2[row]) + b2
  {
    const int tiles = (N / 16) * (OUT_F / 16);
    k_gemm_wmma<HID_F, OUT_F><<<(tiles + 3) / 4, 128, 0, stream>>>(bufB, W2, bufA, N);

    const long elems = (long)N * OUT_F;
    k_self_init<OUT_F><<<(int)((elems + T - 1) / T), T, 0, stream>>>(bufA, dinv, b2, out, N);

    const int wavesPerBlk = T / 32;
    k_edge_scatter<OUT_F><<<(E + wavesPerBlk - 1) / wavesPerBlk, T, 0, stream>>>(
        row, col, dinv, bufA, out, E);
  }
}